// ScorePredictor_10213432230382
// MI455X (gfx1250) — compile-verified
//
#include <hip/hip_runtime.h>

// CDNA5 / gfx1250 wave32 WMMA kernel for ScorePredictor.
//
// pos[e] = sum_p attn_p * raw_p,  raw_p = w_p*(src.news_p) + (1-w_p)*(src.gnn),
// attn = softmax(raw/8), w_p = exp(last_update_p - t0).
// The 9 length-64 FP32 dot products per edge are computed with a chain of
// V_WMMA_F32_16X16X4_F32 ops (A rows = news rows 0..7 + gnn in row 8,
// B = src chunk broadcast across the 16 columns).

typedef __attribute__((ext_vector_type(2))) float v2f;
typedef __attribute__((ext_vector_type(8))) float v8f;

namespace {
constexpr int kC = 8;    // categories
constexpr int kP = 8;    // prefs per category
constexpr int kF = 64;   // feature dim (pref rows are 2*kF wide, first kF used)
}

__global__ __launch_bounds__(256) void score_predictor_wmma(
    const float* __restrict__ src,        // (E, F)
    const int*   __restrict__ cate_id,    // (E,)
    const float* __restrict__ last_upd,   // (E, C, P)
    const float* __restrict__ news_pref,  // (E, C, P, 2F)
    const float* __restrict__ gnn,        // (E, 2F)
    const float* __restrict__ edge_time,  // (E,)  only [0] used
    float* __restrict__ out,              // (E,)
    int E)
{
  const int gwave = (int)((blockIdx.x * (unsigned)blockDim.x + threadIdx.x) >> 5);
  if (gwave >= E) return;                 // wave-uniform: EXEC stays all-1s (WMMA requirement)
  const int e    = gwave;
  const int lane = (int)(threadIdx.x & 31u);
  const int m    = lane & 15;             // A-matrix row this lane feeds
  const int hi   = lane >> 4;             // lane half selects K slots {0,1} vs {2,3}

  const int   c  = cate_id[e];
  const float t0 = edge_time[0];

  // A rows 0..7: news_pref[e, c, p, 0:64]; row 8: gnn[e, 0:64]; rows 9..15 dummy (point at gnn, ignored).
  const float* __restrict__ rowptr =
      (m < kP) ? (news_pref + (((size_t)e * kC + (size_t)c) * kP + (size_t)m) * (2 * kF))
               : (gnn + (size_t)e * (2 * kF));
  const float* __restrict__ srcptr = src + (size_t)e * kF;

  // Two independent accumulation chains (break the D->C dependency chain in half).
  v8f acc0 = {};
  v8f acc1 = {};
#pragma unroll
  for (int j = 0; j < 16; j += 2) {
    const int off0 = 4 * j + 2 * hi;      // K-chunk j, this lane's 2 K-slots
    const int off1 = off0 + 4;            // K-chunk j+1
    v2f a0 = *(const v2f*)(rowptr + off0);
    v2f b0 = *(const v2f*)(srcptr + off0);  // same value within each lane half -> broadcast load
    v2f a1 = *(const v2f*)(rowptr + off1);
    v2f b1 = *(const v2f*)(srcptr + off1);
    // D = A(16x4) x B(4x16) + C ; emits v_wmma_f32_16x16x4_f32
    acc0 = __builtin_amdgcn_wmma_f32_16x16x4_f32(false, a0, false, b0, (short)0, acc0, false, false);
    acc1 = __builtin_amdgcn_wmma_f32_16x16x4_f32(false, a1, false, b1, (short)0, acc1, false, false);
  }
  v8f acc = acc0 + acc1;

  // D layout: VGPR r, lanes 0-15 = row r (cols 0-15, all equal), lanes 16-31 = row r+8.
  float av[kP];
#pragma unroll
  for (int p = 0; p < kP; ++p) av[p] = __shfl(acc[p], 0, 32);   // src . news_p
  const float bdot = __shfl(acc[0], 16, 32);                     // row 8 -> src . gnn

  // last_update[e, c, :] : 32B-aligned, two float4 broadcast loads
  const float* lu = last_upd + ((size_t)e * kC + (size_t)c) * kP;
  const float4 lu0 = ((const float4*)lu)[0];
  const float4 lu1 = ((const float4*)lu)[1];
  const float luv[kP] = {lu0.x, lu0.y, lu0.z, lu0.w, lu1.x, lu1.y, lu1.z, lu1.w};

  float raw[kP];
  float mx = -3.402823466e38f;
#pragma unroll
  for (int p = 0; p < kP; ++p) {
    const float w = __expf(luv[p] - t0);                 // exp(-(t0 - last_update))
    const float r = fmaf(w, av[p] - bdot, bdot);         // w*a + (1-w)*b
    raw[p] = r;
    mx = fmaxf(mx, r);
  }
  // softmax over scores = raw/8 ; pos = sum attn_p * raw_p
  float den = 0.0f, num = 0.0f;
#pragma unroll
  for (int p = 0; p < kP; ++p) {
    const float ep = __expf((raw[p] - mx) * 0.125f);
    den += ep;
    num = fmaf(ep, raw[p], num);
  }
  if (lane == 0) out[e] = num / den;
}

extern "C" void kernel_launch(void* const* d_in, const int* in_sizes, int n_in,
                              void* d_out, int out_size, void* d_ws, size_t ws_size,
                              hipStream_t stream) {
  const float* src  = (const float*)d_in[0];   // src_representation (E,F)
  const int*   cid  = (const int*)  d_in[1];   // cate_id (E,)
  const float* lupd = (const float*)d_in[2];   // dst_last_update (E,C,P)
  const float* np   = (const float*)d_in[3];   // dst_news_pref (E,C,P,2F)
  const float* gnn  = (const float*)d_in[4];   // dst_gnn_emb (E,2F)
  const float* etim = (const float*)d_in[5];   // edge_time (E,)
  float* out = (float*)d_out;

  const int E = in_sizes[1];                   // cate_id flat count == E
  const int wavesPerBlock = 256 / 32;          // 8 waves (one edge each) per block
  const int grid = (E + wavesPerBlock - 1) / wavesPerBlock;
  score_predictor_wmma<<<grid, 256, 0, stream>>>(src, cid, lupd, np, gnn, etim, out, E);
}